// SpectralTopologicalAttention_64630667870451
// MI455X (gfx1250) — compile-verified
//
#include <hip/hip_runtime.h>

// Problem constants (B=2, N=64, D=16)
#define BB 2
#define NN 64
#define DD 16

typedef __attribute__((ext_vector_type(2))) float v2f;
typedef __attribute__((ext_vector_type(8))) float v8f;

__device__ __forceinline__ float wave_sum32(float v) {
#pragma unroll
  for (int off = 16; off > 0; off >>= 1) v += __shfl_xor(v, off, 32);
  return v;
}

__device__ __forceinline__ void lds_fence() {
  asm volatile("s_wait_dscnt 0" ::: "memory");
}

// ---------------------------------------------------------------------------
// Q = X*Wq, K = X*Wk : one thread per (b,n,i)
// ---------------------------------------------------------------------------
__global__ void qk_kernel(const float* __restrict__ X, const float* __restrict__ Wk,
                          const float* __restrict__ Wq, float* __restrict__ Q,
                          float* __restrict__ K) {
  int id = blockIdx.x * blockDim.x + threadIdx.x;  // 2048
  int i = id & 15;
  int bn = id >> 4;
  const float* x = X + bn * 16;
  float q = 0.f, k = 0.f;
#pragma unroll
  for (int d = 0; d < 16; ++d) {
    float xv = x[d];
    q += xv * Wq[d * 16 + i];
    k += xv * Wk[d * 16 + i];
  }
  Q[id] = q;
  K[id] = k;
}

// ---------------------------------------------------------------------------
// V[b,m,s,k,v] = sum_d X[b,m,d] * Wv[d,s,k,v] : one thread per output
// ---------------------------------------------------------------------------
__global__ void v_kernel(const float* __restrict__ X, const float* __restrict__ Wv,
                         float* __restrict__ V) {
  int id = blockIdx.x * blockDim.x + threadIdx.x;  // 196608
  int v = id & 15;
  int k = (id >> 4) & 15;
  int s = (id >> 8) % 6;
  int bm = id / 1536;
  const float* x = X + bm * 16;
  float acc = 0.f;
#pragma unroll
  for (int d = 0; d < 16; ++d)
    acc += x[d] * Wv[((d * 6 + s) * 16 + k) * 16 + v];
  V[id] = acc;
}

// ---------------------------------------------------------------------------
// det(T_k) for k=0..15 via LU with partial pivoting (one thread per k)
// ---------------------------------------------------------------------------
__global__ void dett_kernel(const float* __restrict__ T, float* __restrict__ detT) {
  __shared__ float lds[16 * 257];
  int k = threadIdx.x;  // 0..15
  float* A = lds + k * 257;
  for (int i = 0; i < 16; ++i)
    for (int j = 0; j < 16; ++j)
      A[i * 16 + j] = T[(i * 16 + j) * 16 + k];
  float det = 1.f;
  for (int c = 0; c < 16; ++c) {
    int piv = c;
    float pv = fabsf(A[c * 16 + c]);
    for (int r = c + 1; r < 16; ++r) {
      float av = fabsf(A[r * 16 + c]);
      if (av > pv) { pv = av; piv = r; }
    }
    if (piv != c) {
      det = -det;
      for (int j = c; j < 16; ++j) {
        float t = A[c * 16 + j]; A[c * 16 + j] = A[piv * 16 + j]; A[piv * 16 + j] = t;
      }
    }
    float d = A[c * 16 + c];
    det *= d;
    if (d != 0.f) {
      float invd = 1.f / d;
      for (int r = c + 1; r < 16; ++r) {
        float f = A[r * 16 + c] * invd;
        for (int j = c + 1; j < 16; ++j) A[r * 16 + j] -= f * A[c * 16 + j];
      }
    }
  }
  detT[k] = det;
}

// ---------------------------------------------------------------------------
// H[b,m,k] = (T_k diag(K)) (T_k diag(K))^T : one wave per (b,m,k), WMMA f32.
// For W*W^T the 16x16x4 A-fragment and B-fragment coincide.
// ---------------------------------------------------------------------------
__global__ void h_kernel(const float* __restrict__ T, const float* __restrict__ K,
                         float* __restrict__ H) {
  __shared__ float tile[8][16 * 18];
  __shared__ float kbuf[8][16];
  int lane = threadIdx.x & 31, wid = threadIdx.x >> 5;
  int item = blockIdx.x * 8 + wid;  // b*1024 + m*16 + k, < 2048
  int k = item & 15, m = (item >> 4) & 63, b = item >> 10;
  float* W = tile[wid];
  if (lane < 16) kbuf[wid][lane] = K[(b * 64 + m) * 16 + lane];
  lds_fence();
#pragma unroll
  for (int u = 0; u < 8; ++u) {
    int e = lane * 8 + u, i = e >> 4, j = e & 15;
    W[i * 18 + j] = T[(i * 16 + j) * 16 + k] * kbuf[wid][j];
  }
  lds_fence();
  int lo = lane & 15, hi = lane >> 4;
  v8f c;
#pragma unroll
  for (int r = 0; r < 8; ++r) c[r] = 0.f;
#pragma unroll
  for (int s = 0; s < 4; ++s) {
    int cb = 4 * s + 2 * hi;
    v2f a;
    a.x = W[lo * 18 + cb];
    a.y = W[lo * 18 + cb + 1];
    c = __builtin_amdgcn_wmma_f32_16x16x4_f32(false, a, false, a, (short)0, c, false, false);
  }
  float* h = H + item * 256;
#pragma unroll
  for (int r = 0; r < 8; ++r) h[(r + 8 * hi) * 16 + lo] = c[r];
}

// ---------------------------------------------------------------------------
// rho kernel: one wave per (b,n,m,k). A = T_k * diag(Q.*K); spectral radius
// via 14 normalized squarings, each squaring = 4x v_wmma_f32_16x16x4_f32.
// ---------------------------------------------------------------------------
__global__ void rho_kernel(const float* __restrict__ T, const float* __restrict__ Q,
                           const float* __restrict__ K, float* __restrict__ Sign) {
  __shared__ float tiles[8][16 * 18];
  __shared__ float pbuf[8][16];
  int lane = threadIdx.x & 31, wid = threadIdx.x >> 5;
  int item = blockIdx.x * 8 + wid;  // b*65536 + n*1024 + m*16 + k
  int k = item & 15, m = (item >> 4) & 63, n = (item >> 10) & 63, b = item >> 16;
  float* A = tiles[wid];
  if (lane < 16)
    pbuf[wid][lane] = Q[(b * 64 + n) * 16 + lane] * K[(b * 64 + m) * 16 + lane];
  lds_fence();
#pragma unroll
  for (int u = 0; u < 8; ++u) {
    int e = lane * 8 + u, i = e >> 4, j = e & 15;
    A[i * 18 + j] = T[(i * 16 + j) * 16 + k] * pbuf[wid][j];
  }
  lds_fence();
  int lo = lane & 15, hi = lane >> 4;
  const int ITER = 14;
  float logacc = 0.f, fscale = 1.f;
  for (int it = 0; it <= ITER; ++it) {
    v2f af[4];
    float loc = 0.f;
#pragma unroll
    for (int s = 0; s < 4; ++s) {
      int cb = 4 * s + 2 * hi;
      af[s].x = A[lo * 18 + cb];
      af[s].y = A[lo * 18 + cb + 1];
      loc += af[s].x * af[s].x + af[s].y * af[s].y;
    }
    float nu = sqrtf(wave_sum32(loc));
    logacc += fscale * logf(nu);
    if (it == ITER) break;
    fscale *= 0.5f;
    float inv = (nu > 1e-37f) ? (1.f / nu) : 0.f;
    v2f bf[4];
#pragma unroll
    for (int s = 0; s < 4; ++s) {
      int cb = 4 * s + 2 * hi;
      af[s].x *= inv;
      af[s].y *= inv;
      bf[s].x = A[(cb + 0) * 18 + lo] * inv;
      bf[s].y = A[(cb + 1) * 18 + lo] * inv;
    }
    v8f c;
#pragma unroll
    for (int r = 0; r < 8; ++r) c[r] = 0.f;
#pragma unroll
    for (int s = 0; s < 4; ++s)
      c = __builtin_amdgcn_wmma_f32_16x16x4_f32(false, af[s], false, bf[s], (short)0, c,
                                                false, false);
#pragma unroll
    for (int r = 0; r < 8; ++r) A[(r + 8 * hi) * 18 + lo] = c[r];
    lds_fence();
  }
  float rho = expf(logacc);
  if (lane == 0)
    Sign[(((b * 64 + n) * 64 + m) * 6 + 5) * 16 + k] = rho;
}

// ---------------------------------------------------------------------------
// Jacobi kernel: one thread per (b,n,m,k). S = diag(Q) H diag(Q) symmetric;
// cyclic Jacobi -> sigma_min/max, rank = sum log(sigma+eps); det/trace closed.
// ---------------------------------------------------------------------------
__global__ void jacobi_kernel(const float* __restrict__ Q, const float* __restrict__ K,
                              const float* __restrict__ H, const float* __restrict__ T,
                              const float* __restrict__ detT, float* __restrict__ Sign) {
  __shared__ float lds[32 * 257];
  int tid = threadIdx.x;
  int item = blockIdx.x * 32 + tid;
  int k = item & 15, m = (item >> 4) & 63, n = (item >> 10) & 63, b = item >> 16;
  float* S = lds + tid * 257;
  float q[16], kk[16];
  float prodQ = 1.f, prodK = 1.f, tr = 0.f;
#pragma unroll
  for (int i = 0; i < 16; ++i) {
    q[i] = Q[(b * 64 + n) * 16 + i];
    kk[i] = K[(b * 64 + m) * 16 + i];
    prodQ *= q[i];
    prodK *= kk[i];
    tr += q[i] * kk[i] * T[(i * 16 + i) * 16 + k];
  }
  const float* h = H + ((b * 64 + m) * 16 + k) * 256;
  for (int i = 0; i < 16; ++i)
#pragma unroll
    for (int j = 0; j < 16; ++j)
      S[i * 16 + j] = q[i] * q[j] * h[i * 16 + j];
  // cyclic Jacobi sweeps
  for (int sweep = 0; sweep < 8; ++sweep) {
    for (int p = 0; p < 15; ++p) {
      for (int qq = p + 1; qq < 16; ++qq) {
        float apq = S[p * 16 + qq];
        if (fabsf(apq) < 1e-34f) continue;
        float app = S[p * 16 + p], aqq = S[qq * 16 + qq];
        float theta = (aqq - app) / (2.f * apq);
        float t = 1.f / (fabsf(theta) + sqrtf(theta * theta + 1.f));
        if (theta < 0.f) t = -t;
        float cth = 1.f / sqrtf(t * t + 1.f);
        float sth = t * cth;
        for (int j = 0; j < 16; ++j) {
          float up = S[p * 16 + j], uq = S[qq * 16 + j];
          S[p * 16 + j] = cth * up - sth * uq;
          S[qq * 16 + j] = sth * up + cth * uq;
        }
        for (int i2 = 0; i2 < 16; ++i2) {
          float up = S[i2 * 16 + p], uq = S[i2 * 16 + qq];
          S[i2 * 16 + p] = cth * up - sth * uq;
          S[i2 * 16 + qq] = sth * up + cth * uq;
        }
      }
    }
  }
  float smax = 0.f, smin = 3.4e38f, rank = 0.f;
#pragma unroll
  for (int i = 0; i < 16; ++i) {
    float lam = S[i * 16 + i];
    lam = lam > 0.f ? lam : 0.f;
    float sv = sqrtf(lam);
    smax = fmaxf(smax, sv);
    smin = fminf(smin, sv);
    rank += logf(sv + 1e-6f);
  }
  float det = detT[k] * prodQ * prodK;
  float* sg = Sign + (((b * 64 + n) * 64 + m) * 6) * 16 + k;
  sg[0] = det;        // s=0 det
  sg[16] = tr;        // s=1 trace
  sg[32] = rank;      // s=2 rank_approx
  sg[48] = smin;      // s=3 sigma_min
  sg[64] = smax;      // s=4 sigma_max
}

// ---------------------------------------------------------------------------
// out[b,n,v] = sum_{m,s,k} Sign[b,n,m,s,k] * V[b,m,s,k,v]
// ---------------------------------------------------------------------------
__global__ void combine_kernel(const float* __restrict__ Sign, const float* __restrict__ V,
                               float* __restrict__ out) {
  int id = blockIdx.x * blockDim.x + threadIdx.x;  // 2048
  int v = id & 15;
  int n = (id >> 4) & 63;
  int b = id >> 10;
  const float* sg = Sign + ((b * 64 + n) * 64) * 96;  // 6144 per (b,n)
  const float* vb = V + (b * 64) * 1536;
  float acc = 0.f;
  for (int m = 0; m < 64; ++m) {
    const float* sgm = sg + m * 96;
    const float* vm = vb + m * 1536;
    __builtin_prefetch(vm + 1536, 0, 1);
#pragma unroll 4
    for (int t = 0; t < 96; ++t)  // t = s*16 + k
      acc += sgm[t] * vm[t * 16 + v];
  }
  out[id] = acc;
}

// ---------------------------------------------------------------------------
extern "C" void kernel_launch(void* const* d_in, const int* in_sizes, int n_in,
                              void* d_out, int out_size, void* d_ws, size_t ws_size,
                              hipStream_t stream) {
  const float* X  = (const float*)d_in[0];
  const float* Wk = (const float*)d_in[1];
  const float* Wq = (const float*)d_in[2];
  const float* T  = (const float*)d_in[3];
  const float* Wv = (const float*)d_in[4];
  float* out = (float*)d_out;

  float* W = (float*)d_ws;
  float* Q    = W;             // 2048
  float* K    = W + 2048;      // 2048
  float* detT = W + 4096;      // 16
  float* V    = W + 4112;      // 196608
  float* H    = W + 200720;    // 524288
  float* Sign = W + 725008;    // 786432  (total ~5.8 MB)

  qk_kernel<<<8, 256, 0, stream>>>(X, Wk, Wq, Q, K);
  v_kernel<<<768, 256, 0, stream>>>(X, Wv, V);
  dett_kernel<<<1, 16, 0, stream>>>(T, detT);
  h_kernel<<<256, 256, 0, stream>>>(T, K, H);
  rho_kernel<<<16384, 256, 0, stream>>>(T, Q, K, Sign);
  jacobi_kernel<<<4096, 32, 0, stream>>>(Q, K, H, T, detT, Sign);
  combine_kernel<<<8, 256, 0, stream>>>(Sign, V, out);
}